// GCN_13683765805327
// MI455X (gfx1250) — compile-verified
//
#include <hip/hip_runtime.h>
#include <hip/hip_bf16.h>

// ---------------------------------------------------------------------------
// GCN (3x GCNConv + BN + ReLU + log_softmax), CDNA5 / gfx1250.
// GEMMs via v_wmma_f32_16x16x4_f32 with software-pipelined (double-buffered)
// A/B register fragments. Edge aggregation via L2-resident gather +
// global_atomic_add_f32. Layer-1 X stream tagged non-temporal to preserve L2
// for the random-access scatter phases.
// ---------------------------------------------------------------------------

typedef __attribute__((ext_vector_type(2))) float v2f;
typedef __attribute__((ext_vector_type(8))) float v8f;

#define NNODES_FDIM 512

__device__ __forceinline__ void atomic_fadd(float* p, float v) {
    // hardware FP32 atomic add (global_atomic_add_f32), no CAS loop
    unsafeAtomicAdd(p, v);
}

// ---------------- degree / norm ----------------

__global__ void k_deg_init(float* deg, int n) {
    int i = blockIdx.x * blockDim.x + threadIdx.x;
    if (i < n) deg[i] = 1.0f;   // self-loop contributes 1
}

__global__ void k_deg_count(const long long* __restrict__ ei, float* deg, int E_) {
    int e = blockIdx.x * blockDim.x + threadIdx.x;
    if (e < E_) {
        int d = (int)ei[(size_t)E_ + e];
        atomic_fadd(&deg[d], 1.0f);
    }
}

__global__ void k_rsqrt_inplace(float* d, int n) {
    int i = blockIdx.x * blockDim.x + threadIdx.x;
    if (i < n) d[i] = rsqrtf(d[i]);
}

__global__ void k_edge_norm(const long long* __restrict__ ei,
                            const float* __restrict__ dinv,
                            float* __restrict__ norm, int E_) {
    int e = blockIdx.x * blockDim.x + threadIdx.x;
    if (e < E_) {
        int s = (int)ei[e];
        int d = (int)ei[(size_t)E_ + e];
        norm[e] = dinv[s] * dinv[d];
    }
}

// ---------------- WMMA GEMM: Y[n x M] = X[n x K] @ W[K x M] ----------------
// One wave computes a 16-row strip across all MT 16-wide column tiles, so X
// is streamed from memory exactly once (layer 1: 205 MB HBM, one NT pass).
// A/B fragments are double-buffered so next k-step loads overlap current
// WMMAs instead of each WMMA stalling on s_wait_loadcnt 0.

template<int K, int MT, bool NT_A>
__global__ void k_gemm_wmma_f32(const float* __restrict__ X,
                                const float* __restrict__ W,
                                float* __restrict__ Y,
                                int n_rows, int M) {
    const int lane    = threadIdx.x & 31;
    const int wave    = threadIdx.x >> 5;
    const int tile    = blockIdx.x * (blockDim.x >> 5) + wave;
    const int n_tiles = n_rows >> 4;
    if (tile >= n_tiles) return;          // uniform per wave: EXEC stays full

    const int halfsel = lane >> 4;        // 0: K pair {0,1}, 1: K pair {2,3}
    const int l16     = lane & 15;
    const int row0    = tile << 4;
    const float* xrow = X + (size_t)(row0 + l16) * K;

    v8f acc[MT];
    v8f zero = {};
#pragma unroll
    for (int t = 0; t < MT; ++t) acc[t] = zero;

    v2f a_cur = {}, a_nxt = {};
    v2f b_cur[MT], b_nxt[MT];
#pragma unroll
    for (int t = 0; t < MT; ++t) { b_cur[t] = a_cur; b_nxt[t] = a_cur; }

    auto load_ab = [&](int k0, v2f& a, v2f* b) {
        const int ka = k0 + (halfsel << 1);
        if (NT_A) {
            a = __builtin_nontemporal_load(
                    reinterpret_cast<const v2f*>(xrow + ka));
        } else {
            a = *reinterpret_cast<const v2f*>(xrow + ka);
        }
#pragma unroll
        for (int t = 0; t < MT; ++t) {
            const int col = t * 16 + l16;
            if (MT * 16 == 64 || MT * 16 == 128 || col < M) {
                b[t].x = W[(size_t)ka * M + col];
                b[t].y = W[(size_t)(ka + 1) * M + col];
            } else {
                b[t].x = 0.0f; b[t].y = 0.0f;   // zero-pad cols (layer 3, M=40)
            }
        }
    };

    load_ab(0, a_cur, b_cur);
#pragma unroll 2
    for (int k0 = 0; k0 < K; k0 += 4) {
        if (k0 + 4 < K) load_ab(k0 + 4, a_nxt, b_nxt);   // prefetch next step
#pragma unroll
        for (int t = 0; t < MT; ++t) {
            acc[t] = __builtin_amdgcn_wmma_f32_16x16x4_f32(
                false, a_cur, false, b_cur[t], (short)0, acc[t], false, false);
        }
        a_cur = a_nxt;
#pragma unroll
        for (int t = 0; t < MT; ++t) b_cur[t] = b_nxt[t];
    }

    // C/D layout: VGPR r -> row r (lanes 0-15) / row r+8 (lanes 16-31)
#pragma unroll
    for (int t = 0; t < MT; ++t) {
        const int col = t * 16 + l16;
        if (col < M) {
#pragma unroll
            for (int r = 0; r < 8; ++r) {
                const int orow = row0 + (halfsel << 3) + r;
                Y[(size_t)orow * M + col] = acc[t][r];
            }
        }
    }
}

// ---------------- aggregation ----------------

// agg[i] = h[i] * dinv[i]^2   (self-loop term; also fully initializes agg)
template<int H>
__global__ void k_self_init(const float* __restrict__ h,
                            const float* __restrict__ dinv,
                            float* __restrict__ agg, int n) {
    const int H4 = H / 4;
    long long t = (long long)blockIdx.x * blockDim.x + threadIdx.x;
    if (t >= (long long)n * H4) return;
    int i = (int)(t / H4), c = (int)(t % H4);
    float w = dinv[i]; w *= w;
    float4 v = *reinterpret_cast<const float4*>(h + (size_t)i * H + c * 4);
    float4 o; o.x = v.x * w; o.y = v.y * w; o.z = v.z * w; o.w = v.w * w;
    *reinterpret_cast<float4*>(agg + (size_t)i * H + c * 4) = o;
}

// agg[dst] += norm[e] * h[src]   (L2-resident random access)
template<int H>
__global__ void k_scatter_edges(const long long* __restrict__ ei,
                                const float* __restrict__ norm,
                                const float* __restrict__ h,
                                float* __restrict__ agg, int E_) {
    const int H4 = H / 4;
    long long t = (long long)blockIdx.x * blockDim.x + threadIdx.x;
    if (t >= (long long)E_ * H4) return;
    int e = (int)(t / H4), c = (int)(t % H4);
    int s = (int)ei[e];
    int d = (int)ei[(size_t)E_ + e];
    float w = norm[e];
    float4 v = *reinterpret_cast<const float4*>(h + (size_t)s * H + c * 4);
    float* o = agg + (size_t)d * H + c * 4;
    atomic_fadd(o + 0, v.x * w);
    atomic_fadd(o + 1, v.y * w);
    atomic_fadd(o + 2, v.z * w);
    atomic_fadd(o + 3, v.w * w);
}

// ---------------- fused bias + BN(eval) + ReLU ----------------

// scale = g / sqrt(rv + eps); shift = (b - rm) * scale + bt
__global__ void k_bn_params(const float* __restrict__ b, const float* __restrict__ g,
                            const float* __restrict__ bt, const float* __restrict__ rm,
                            const float* __restrict__ rv,
                            float* __restrict__ scale, float* __restrict__ shift, int H) {
    int f = blockIdx.x * blockDim.x + threadIdx.x;
    if (f < H) {
        float s = g[f] * rsqrtf(rv[f] + 1e-5f);
        scale[f] = s;
        shift[f] = (b[f] - rm[f]) * s + bt[f];
    }
}

template<int H>
__global__ void k_bn_relu(const float* __restrict__ agg,
                          const float* __restrict__ scale,
                          const float* __restrict__ shift,
                          float* __restrict__ y, int n) {
    const int H4 = H / 4;
    long long t = (long long)blockIdx.x * blockDim.x + threadIdx.x;
    if (t >= (long long)n * H4) return;
    int i = (int)(t / H4), c = (int)(t % H4);
    float4 a  = *reinterpret_cast<const float4*>(agg   + (size_t)i * H + c * 4);
    float4 s  = *reinterpret_cast<const float4*>(scale + c * 4);
    float4 sh = *reinterpret_cast<const float4*>(shift + c * 4);
    float4 o;
    o.x = fmaxf(fmaf(a.x, s.x, sh.x), 0.0f);
    o.y = fmaxf(fmaf(a.y, s.y, sh.y), 0.0f);
    o.z = fmaxf(fmaf(a.z, s.z, sh.z), 0.0f);
    o.w = fmaxf(fmaf(a.w, s.w, sh.w), 0.0f);
    *reinterpret_cast<float4*>(y + (size_t)i * H + c * 4) = o;
}

// ---------------- final: +b3, log_softmax over C=40 ----------------

__global__ void k_final_lsm(const float* __restrict__ agg,
                            const float* __restrict__ b3,
                            float* __restrict__ out, int n) {
    int i = blockIdx.x * blockDim.x + threadIdx.x;
    if (i >= n) return;
    float v[40];
    float mx = -1e30f;
#pragma unroll
    for (int c = 0; c < 40; ++c) {
        v[c] = agg[(size_t)i * 40 + c] + b3[c];
        mx = fmaxf(mx, v[c]);
    }
    float s = 0.0f;
#pragma unroll
    for (int c = 0; c < 40; ++c) s += expf(v[c] - mx);
    float lse = mx + logf(s);
#pragma unroll
    for (int c = 0; c < 40; ++c) out[(size_t)i * 40 + c] = v[c] - lse;
}

// ---------------- host ----------------

static inline int gb(long long total, int bs) { return (int)((total + bs - 1) / bs); }
static inline size_t align_up(size_t x, size_t a) { return (x + a - 1) / a * a; }

extern "C" void kernel_launch(void* const* d_in, const int* in_sizes, int n_in,
                              void* d_out, int out_size, void* d_ws, size_t ws_size,
                              hipStream_t stream) {
    const float*     x   = (const float*)d_in[0];
    const long long* ei  = (const long long*)d_in[1];
    const float*     W1  = (const float*)d_in[2];
    const float*     b1  = (const float*)d_in[3];
    const float*     W2  = (const float*)d_in[4];
    const float*     b2  = (const float*)d_in[5];
    const float*     W3  = (const float*)d_in[6];
    const float*     b3  = (const float*)d_in[7];
    const float*     g1  = (const float*)d_in[8];
    const float*     bt1 = (const float*)d_in[9];
    const float*     rm1 = (const float*)d_in[10];
    const float*     rv1 = (const float*)d_in[11];
    const float*     g2  = (const float*)d_in[12];
    const float*     bt2 = (const float*)d_in[13];
    const float*     rm2 = (const float*)d_in[14];
    const float*     rv2 = (const float*)d_in[15];

    const int N = in_sizes[0] / NNODES_FDIM;   // 100000
    const int E = in_sizes[1] / 2;             // 3200000

    // workspace layout
    char* ws = (char*)d_ws;
    size_t off = 0;
    auto alloc = [&](size_t bytes) -> void* {
        void* p = ws + off;
        off = align_up(off + bytes, 256);
        return p;
    };
    float* prm  = (float*)alloc(512 * sizeof(float));          // scale1|shift1|scale2|shift2
    float* dinv = (float*)alloc((size_t)N * sizeof(float));    // deg -> dinv in place
    float* norm = (float*)alloc((size_t)E * sizeof(float));
    float* P0   = (float*)alloc((size_t)N * 64  * sizeof(float));
    float* P1   = (float*)alloc((size_t)N * 128 * sizeof(float));
    float* P2   = (float*)alloc((size_t)N * 128 * sizeof(float));
    float* scale1 = prm + 0,   *shift1 = prm + 128;
    float* scale2 = prm + 256, *shift2 = prm + 384;

    const int BS = 256;
    const int gemm_blocks = gb(N / 16, 8);   // 8 waves/block, 1 row-tile/wave

    // degrees, dinv, per-edge norm
    k_deg_init<<<gb(N, BS), BS, 0, stream>>>(dinv, N);
    k_deg_count<<<gb(E, BS), BS, 0, stream>>>(ei, dinv, E);
    k_rsqrt_inplace<<<gb(N, BS), BS, 0, stream>>>(dinv, N);
    k_edge_norm<<<gb(E, BS), BS, 0, stream>>>(ei, dinv, norm, E);
    k_bn_params<<<1, 128, 0, stream>>>(b1, g1, bt1, rm1, rv1, scale1, shift1, 64);
    k_bn_params<<<1, 128, 0, stream>>>(b2, g2, bt2, rm2, rv2, scale2, shift2, 128);

    // ---- layer 1: 512 -> 64 ----
    k_gemm_wmma_f32<512, 4, true><<<gemm_blocks, BS, 0, stream>>>(x, W1, P1, N, 64);
    k_self_init<64><<<gb((long long)N * 16, BS), BS, 0, stream>>>(P1, dinv, P2, N);
    k_scatter_edges<64><<<gb((long long)E * 16, BS), BS, 0, stream>>>(ei, norm, P1, P2, E);
    k_bn_relu<64><<<gb((long long)N * 16, BS), BS, 0, stream>>>(P2, scale1, shift1, P0, N);

    // ---- layer 2: 64 -> 128 ----
    k_gemm_wmma_f32<64, 8, false><<<gemm_blocks, BS, 0, stream>>>(P0, W2, P1, N, 128);
    k_self_init<128><<<gb((long long)N * 32, BS), BS, 0, stream>>>(P1, dinv, P2, N);
    k_scatter_edges<128><<<gb((long long)E * 32, BS), BS, 0, stream>>>(ei, norm, P1, P2, E);
    k_bn_relu<128><<<gb((long long)N * 32, BS), BS, 0, stream>>>(P2, scale2, shift2, P1, N);

    // ---- layer 3: 128 -> 40 (padded to 48 cols inside the GEMM) ----
    k_gemm_wmma_f32<128, 3, false><<<gemm_blocks, BS, 0, stream>>>(P1, W3, P0, N, 40);
    k_self_init<40><<<gb((long long)N * 10, BS), BS, 0, stream>>>(P0, dinv, P2, N);
    k_scatter_edges<40><<<gb((long long)E * 10, BS), BS, 0, stream>>>(ei, norm, P0, P2, E);

    // ---- bias + log_softmax ----
    k_final_lsm<<<gb(N, BS), BS, 0, stream>>>(P2, b3, (float*)d_out, N);
}